// LSTM_21019569947052
// MI455X (gfx1250) — compile-verified
//
#include <hip/hip_runtime.h>

// Problem constants (match reference)
#define BB 128   // batch
#define SS 256   // seq len
#define II 512   // input dim
#define HH 768   // hidden
#define G4 3072  // 4*H

typedef __attribute__((ext_vector_type(2))) float v2f;
typedef __attribute__((ext_vector_type(8))) float v8f;

// CDNA5 fp32 WMMA: D(16x16) = A(16x4) * B(4x16) + C
__device__ __forceinline__ v8f wmma_f32(v2f a, v2f b, v8f c) {
    return __builtin_amdgcn_wmma_f32_16x16x4_f32(
        /*neg_a=*/false, a, /*neg_b=*/false, b,
        /*c_mod=*/(short)0, c, /*reuse_a=*/false, /*reuse_b=*/false);
}

__device__ __forceinline__ float sigm(float x) {
    return 1.0f / (1.0f + __expf(-x));               // exp(+inf) -> 0, no NaN
}
__device__ __forceinline__ float tanh_fast(float x) {
    return 1.0f - 2.0f / (1.0f + __expf(2.0f * x));  // saturates to +/-1, no NaN
}

__global__ void zero_kernel(float* __restrict__ p, int n) {
    int i = blockIdx.x * blockDim.x + threadIdx.x;
    if (i < n) p[i] = 0.0f;
}

// xg[b*S+s, :] = X[b*S + (flip? S-1-s : s), :K] @ Wx[K, 4H] + bx
// Block: 128 threads (4 waves), tile BM=32 x BN=64, K-chunk 32.
// B tile stored TRANSPOSED in LDS (sBt[n][kk], stride 34) so each WMMA B
// fragment is one contiguous conflict-free ds_load_b64.
__global__ __launch_bounds__(128) void gemm_xg(const float* __restrict__ X,
                                               const float* __restrict__ Wx,
                                               const float* __restrict__ bx,
                                               float* __restrict__ xg,
                                               int K, int flip) {
    __shared__ float sA[32][34];   // [m][kk]
    __shared__ float sBt[64][34];  // [n][kk]  (transposed)
    const int n0   = blockIdx.x * 64;
    const int m0   = blockIdx.y * 32;
    const int tid  = threadIdx.x;
    const int lane = tid & 31;
    const int wid  = tid >> 5;
    const int wm   = (wid >> 1) * 16;  // 0 or 16 (rows)
    const int wn   = (wid & 1) * 32;   // 0 or 32 (cols, 2x 16-wide accs)
    const int fm   = lane & 15;        // M (A) / N (B) index in fragment
    const int fk   = (lane >> 4) * 2;  // K pair selector

    v8f acc0 = {}; v8f acc1 = {};

    for (int k0 = 0; k0 < K; k0 += 32) {
        // A tile: rows coalesced, conflict-free stores
        for (int idx = tid; idx < 32 * 32; idx += 128) {
            int m = idx >> 5, kk = idx & 31;
            int gm = m0 + m;
            int srow = gm;
            if (flip) { int b = gm >> 8; int s = gm & 255; srow = (b << 8) + (255 - s); }
            sA[m][kk] = X[(long)srow * K + k0 + kk];
        }
        // B tile: global-coalesced in n, stored transposed (conflict-free: 34n mod 64 distinct)
        for (int idx = tid; idx < 32 * 64; idx += 128) {
            int kk = idx >> 6, n = idx & 63;
            sBt[n][kk] = Wx[(long)(k0 + kk) * G4 + n0 + n];
        }
        __syncthreads();
#pragma unroll
        for (int kk = 0; kk < 32; kk += 4) {
            v2f a  = *(const v2f*)&sA[wm + fm][kk + fk];
            v2f b0 = *(const v2f*)&sBt[wn + fm][kk + fk];
            v2f b1 = *(const v2f*)&sBt[wn + 16 + fm][kk + fk];
            acc0 = wmma_f32(a, b0, acc0);
            acc1 = wmma_f32(a, b1, acc1);
        }
        __syncthreads();
    }
    const int nloc = lane & 15;
    const int mhi  = (lane >> 4) * 8;
#pragma unroll
    for (int r = 0; r < 8; ++r) {
        long gm  = m0 + wm + mhi + r;
        int  gn0 = n0 + wn + nloc;
        int  gn1 = gn0 + 16;
        xg[gm * G4 + gn0] = acc0[r] + bx[gn0];
        xg[gm * G4 + gn1] = acc1[r] + bx[gn1];
    }
}

// One LSTM time step, fully fused: gates = xg[:,t,:] + h_prev @ Wh + bh,
// then cell update; writes h_next, c_next and the output sequence slot.
// Block: 128 threads (4 waves), tile 32 batch x 32 hcols, all 4 gates.
// Wh tiles stored transposed in LDS for single-b64 B fragments.
__global__ __launch_bounds__(128) void lstm_step(const float* __restrict__ xg, int t,
                                                 const float* __restrict__ Wh,
                                                 const float* __restrict__ bh,
                                                 const float* __restrict__ h_prev,
                                                 const float* __restrict__ c_prev,
                                                 float* __restrict__ h_next,
                                                 float* __restrict__ c_next,
                                                 float* __restrict__ out_seq,
                                                 int out_t, int out_w, int out_coff) {
    __shared__ float sH[32][34];        // [m][kk]
    __shared__ float sWt[4][32][34];    // [gate][n][kk] (transposed)
    const int nc0  = blockIdx.x * 32;   // hidden-col tile
    const int m0   = blockIdx.y * 32;   // batch tile
    const int tid  = threadIdx.x;
    const int lane = tid & 31;
    const int wid  = tid >> 5;
    const int wm   = (wid >> 1) * 16;
    const int wn   = (wid & 1) * 16;
    const int fm   = lane & 15;
    const int fk   = (lane >> 4) * 2;

    v8f acc[4] = {};

    for (int k0 = 0; k0 < HH; k0 += 32) {
        for (int idx = tid; idx < 32 * 32; idx += 128) {
            int m = idx >> 5, kk = idx & 31;
            sH[m][kk] = h_prev[(m0 + m) * HH + k0 + kk];
        }
        // 4 gate tiles, global-coalesced in n, stored transposed
        for (int idx = tid; idx < 4 * 32 * 32; idx += 128) {
            int g = idx >> 10, rem = idx & 1023;
            int kk = rem >> 5, n = rem & 31;
            sWt[g][n][kk] = Wh[(long)(k0 + kk) * G4 + g * HH + nc0 + n];
        }
        __syncthreads();
#pragma unroll
        for (int kk = 0; kk < 32; kk += 4) {
            v2f a = *(const v2f*)&sH[wm + fm][kk + fk];
#pragma unroll
            for (int g = 0; g < 4; ++g) {   // reuse A fragment across the 4 gates
                v2f b = *(const v2f*)&sWt[g][wn + fm][kk + fk];
                acc[g] = wmma_f32(a, b, acc[g]);
            }
        }
        __syncthreads();
    }

    const int nloc = lane & 15;
    const int mhi  = (lane >> 4) * 8;
#pragma unroll
    for (int r = 0; r < 8; ++r) {
        int bi = m0 + wm + mhi + r;
        int hc = nc0 + wn + nloc;
        long xb = ((long)bi * SS + t) * G4 + hc;
        float gf = acc[0][r] + xg[xb]            + bh[hc];
        float gi = acc[1][r] + xg[xb + HH]       + bh[hc + HH];
        float gg = acc[2][r] + xg[xb + 2 * HH]   + bh[hc + 2 * HH];
        float go = acc[3][r] + xg[xb + 3 * HH]   + bh[hc + 3 * HH];
        float f = sigm(gf), i = sigm(gi), g = tanh_fast(gg), o = sigm(go);
        float c = f * c_prev[bi * HH + hc] + i * g;
        float h = o * tanh_fast(c);
        c_next[bi * HH + hc] = c;
        h_next[bi * HH + hc] = h;
        out_seq[((long)bi * SS + out_t) * out_w + out_coff + hc] = h;
    }
}

extern "C" void kernel_launch(void* const* d_in, const int* in_sizes, int n_in,
                              void* d_out, int out_size, void* d_ws, size_t ws_size,
                              hipStream_t stream) {
    const float* x = (const float*)d_in[0];
    // Input tag order: f0, f1, b0, b1; each tag = {Wx, bx, Wh, bh}
    const float *Wx[4], *bx[4], *Wh[4], *bh[4];
    for (int tg = 0; tg < 4; ++tg) {
        Wx[tg] = (const float*)d_in[1 + 4 * tg + 0];
        bx[tg] = (const float*)d_in[1 + 4 * tg + 1];
        Wh[tg] = (const float*)d_in[1 + 4 * tg + 2];
        bh[tg] = (const float*)d_in[1 + 4 * tg + 3];
    }
    float* out = (float*)d_out;                        // [B,S,2H]
    float* hn  = out + (long)BB * SS * 2 * HH;         // [4,B,H] : f0,f1,b0,b1
    float* cn  = hn + 4L * BB * HH;                    // [4,B,H]

    float* ws   = (float*)d_ws;
    float* xg   = ws;                                  // [B,S,4H]
    float* seqF = xg + (long)BB * SS * G4;             // [B,S,H] layer0 fwd out
    float* seqB = seqF + (long)BB * SS * HH;           // [B,S,H] layer0 bwd out (reversed order)
    float* hA   = seqB + (long)BB * SS * HH;           // ping-pong h/c
    float* cA   = hA + BB * HH;
    float* hB   = cA + BB * HH;
    float* cB   = hB + BB * HH;
    float* hbuf[2] = {hA, hB};
    float* cbuf[2] = {cA, cB};

    struct Pass {
        const float* src; int K; int flip_in; int tag;
        float* oseq; int ow; int coff; int flip_out; int hidx;
    };
    Pass passes[4] = {
        { x,    II, 0, 0, seqF, HH,     0,  0, 0 },  // fwd layer0 -> h_n[0]
        { x,    II, 1, 2, seqB, HH,     0,  0, 2 },  // bwd layer0 -> h_n[2]
        { seqF, HH, 0, 1, out,  2 * HH, 0,  0, 1 },  // fwd layer1 -> out[:, :, 0:H],  h_n[1]
        { seqB, HH, 0, 3, out,  2 * HH, HH, 1, 3 },  // bwd layer1 -> out[:, :, H:2H] (flipped), h_n[3]
    };

    for (int p = 0; p < 4; ++p) {
        const Pass& P = passes[p];
        // Parallel input-projection GEMM (hoisted out of the scan)
        dim3 gg(G4 / 64, (BB * SS) / 32);
        gemm_xg<<<gg, 128, 0, stream>>>(P.src, Wx[P.tag], bx[P.tag], xg, P.K, P.flip_in);
        // h0 = c0 = 0 (hA, cA are contiguous)
        int nz = 2 * BB * HH;
        zero_kernel<<<(nz + 255) / 256, 256, 0, stream>>>(hA, nz);
        // Sequential scan: one fused GEMM+cell kernel per step, ping-pong h/c
        int cur = 0;
        for (int t = 0; t < SS; ++t) {
            float* hnx = (t == SS - 1) ? (hn + (long)P.hidx * BB * HH) : hbuf[1 - cur];
            float* cnx = (t == SS - 1) ? (cn + (long)P.hidx * BB * HH) : cbuf[1 - cur];
            int ot = P.flip_out ? (SS - 1 - t) : t;
            dim3 gs(HH / 32, BB / 32);
            lstm_step<<<gs, 128, 0, stream>>>(xg, t, Wh[P.tag], bh[P.tag],
                                              hbuf[cur], cbuf[cur], hnx, cnx,
                                              P.oseq, ot, P.ow, P.coff);
            cur ^= 1;
        }
    }
}